// DeepHitLossWithBounds_21036749815841
// MI455X (gfx1250) — compile-verified
//
#include <hip/hip_runtime.h>
#include <hip/hip_bf16.h>

typedef float v2f __attribute__((ext_vector_type(2)));
typedef float v8f __attribute__((ext_vector_type(8)));

#define T_DIM        64
#define EPSF         1e-10f
#define RANK_MARGIN  0.1f
#define BOUND_MARGIN 0.05f
#define K2_BLOCKS    1024
#define JBLK         4          // j-tiles (16 cols each) per wave task

__device__ __forceinline__ float pos_inf() { return __int_as_float(0x7f800000); }

// ---------------------------------------------------------------------------
// Kernel 1: per-row stats. One thread per sample row.
//   C[i,k] = cumsum_k risk_probs[i,:]          (workspace, feeds WMMA B-operand)
//   P[i]   = { ev? t : +INF, RANK_M - Ci, BOUND_M - Ri, b_i }   (i-side pack)
//   Q[j]   = { t_j, ev_j, R_j, b_j }                            (j-side pack)
//   nll_i  = -log(p[i,t_i]+eps)*ev - log(prod_k(1-p+eps)+eps)*(1-ev)
// Block partials, fixed-order reductions (deterministic, no float atomics).
// ---------------------------------------------------------------------------
__global__ __launch_bounds__(256) void k1_rowstats(
    const float* __restrict__ rp, const int* __restrict__ times,
    const int* __restrict__ events, const float* __restrict__ bnd,
    float* __restrict__ C, float4* __restrict__ P, float4* __restrict__ Q,
    float* __restrict__ nllPart, float* __restrict__ mPart, int B)
{
    int i = blockIdx.x * blockDim.x + threadIdx.x;
    float nll_i = 0.0f, mc = 0.0f;
    if (i < B) {
        const float* row  = rp + (size_t)i * T_DIM;
        float*       crow = C  + (size_t)i * T_DIM;
        int   ti   = times[i];
        float ev   = (float)events[i];
        float bi   = bnd[i];
        float csum = 0.0f, surv = 1.0f, p_t = 0.0f, c_t = 0.0f;
        #pragma unroll 8
        for (int k = 0; k < T_DIM; ++k) {
            float p = row[k];
            csum += p;
            crow[k] = csum;
            surv *= (1.0f - p + EPSF);
            bool hit = (k == ti);
            p_t = hit ? p    : p_t;
            c_t = hit ? csum : c_t;
        }
        // i-side pack: +INF time for censored i kills all its ranking pairs.
        P[i] = make_float4(ev > 0.0f ? (float)ti : pos_inf(),
                           RANK_MARGIN - c_t,
                           BOUND_MARGIN - csum,
                           bi);
        Q[i] = make_float4((float)ti, ev, csum, bi);
        nll_i = -logf(p_t + EPSF) * ev - logf(surv + EPSF) * (1.0f - ev);
        mc    = (bi > 0.0f) ? 1.0f : 0.0f;
    }
    for (int off = 16; off; off >>= 1) {
        nll_i += __shfl_xor(nll_i, off);
        mc    += __shfl_xor(mc, off);
    }
    __shared__ float sN[8], sM[8];
    int lane = threadIdx.x & 31, wid = threadIdx.x >> 5;
    if (lane == 0) { sN[wid] = nll_i; sM[wid] = mc; }
    __syncthreads();
    if (threadIdx.x == 0) {
        float a = 0.0f, b = 0.0f;
        for (int w = 0; w < 8; ++w) { a += sN[w]; b += sM[w]; }
        nllPart[blockIdx.x] = a;
        mPart[blockIdx.x]   = b;
    }
}

// ---------------------------------------------------------------------------
// Kernel 2: pairwise ranking + boundary losses.
// One wave32 owns a 16 x (16*JBLK) strip of the BxB pair matrix.
// A[i,j] = C[j, t_i] is computed EXACTLY via f32 WMMA as E x C^T with a
// one-hot E built in registers from `times` (0/1 weights -> bit-exact).
// The one-hot A-operand and the i-side P loads are shared across JBLK
// accumulator tiles, giving JBLK independent WMMA chains per wave.
// ---------------------------------------------------------------------------
__global__ __launch_bounds__(256) void k2_pairwise(
    const float* __restrict__ C, const int* __restrict__ times,
    const float4* __restrict__ P, const float4* __restrict__ Q,
    float* __restrict__ rankPart, float* __restrict__ npPart,
    float* __restrict__ bsPart, int B)
{
    const int NT         = B >> 4;             // 16-col tiles per dim
    const int NTJ        = NT / JBLK;          // j strip count
    const int totalTasks = NT * NTJ;
    const int lane       = threadIdx.x & 31;
    const int waveId     = threadIdx.x >> 5;
    const int wavesPerBlk= blockDim.x >> 5;
    const int globalWave = blockIdx.x * wavesPerBlk + waveId;
    const int waveStride = gridDim.x * wavesPerBlk;
    const int half       = lane >> 4;          // 0: lanes 0-15, 1: lanes 16-31
    const int nl         = lane & 15;

    float rank_acc = 0.0f, np_acc = 0.0f, bs_acc = 0.0f;

    for (int task = globalWave; task < totalTasks; task += waveStride) {
        const int it    = task / NTJ;
        const int jt    = task - it * NTJ;
        const int iBase = it << 4;
        const int jBase = jt * (16 * JBLK);

        // A-operand: one-hot row M = lane&15 of E; half selects K sub-pair.
        const int tA = times[iBase + nl];

        // B-operand row pointers: lane nl supplies column jBase + u*16 + nl.
        const float* crow[JBLK];
        #pragma unroll
        for (int u = 0; u < JBLK; ++u)
            crow[u] = C + (size_t)(jBase + (u << 4) + nl) * T_DIM + (half << 1);

        // Speculative prefetch of the next task's first C rows.
        {
            int ntask = task + waveStride;
            if (ntask >= totalTasks) ntask = task;
            int njt = ntask - (ntask / NTJ) * NTJ;
            __builtin_prefetch(C + (size_t)(njt * (16 * JBLK) + nl) * T_DIM, 0, 0);
        }

        v8f acc[JBLK];
        #pragma unroll
        for (int u = 0; u < JBLK; ++u)
            acc[u] = (v8f){0.f, 0.f, 0.f, 0.f, 0.f, 0.f, 0.f, 0.f};

        #pragma unroll
        for (int kk = 0; kk < T_DIM; kk += 4) {
            v2f a;
            const int k0 = kk + (half << 1);
            a.x = (tA == k0)     ? 1.0f : 0.0f;
            a.y = (tA == k0 + 1) ? 1.0f : 0.0f;
            #pragma unroll
            for (int u = 0; u < JBLK; ++u) {
                v2f bb = *(const v2f*)(crow[u] + kk);
                acc[u] = __builtin_amdgcn_wmma_f32_16x16x4_f32(
                             false, a, false, bb, (short)0, acc[u], false, false);
            }
        }
        // acc[u][v] = C[jBase + u*16 + nl, t_{iBase + v + 8*half}]  (exact)

        float4 Qv[JBLK];
        #pragma unroll
        for (int u = 0; u < JBLK; ++u)
            Qv[u] = Q[jBase + (u << 4) + nl];   // {t_j, ev_j, R_j, b_j}

        #pragma unroll
        for (int v = 0; v < 8; ++v) {
            const float4 Pv = P[iBase + v + (half << 3)]; // {tMask, mR-Ci, mB-Ri, b_i}
            #pragma unroll
            for (int u = 0; u < JBLK; ++u) {
                // ranking: w = ev_i*ev_j*(t_j > t_i) via +INF fold in Pv.x
                const float w = (Qv[u].x > Pv.x) ? Qv[u].y : 0.0f;
                rank_acc += fmaxf(acc[u][v] + Pv.y, 0.0f) * w;
                np_acc   += w;
                // boundary: (b_j>0 && b_i>b_j) implies b_i>0
                const float bm = ((Qv[u].w > 0.0f) && (Pv.w > Qv[u].w)) ? 1.0f : 0.0f;
                bs_acc   += fmaxf(Qv[u].z + Pv.z, 0.0f) * bm;
            }
        }
    }

    for (int off = 16; off; off >>= 1) {
        rank_acc += __shfl_xor(rank_acc, off);
        np_acc   += __shfl_xor(np_acc, off);
        bs_acc   += __shfl_xor(bs_acc, off);
    }
    __shared__ float sR[8], sP[8], sB[8];
    if (lane == 0) { sR[waveId] = rank_acc; sP[waveId] = np_acc; sB[waveId] = bs_acc; }
    __syncthreads();
    if (threadIdx.x == 0) {
        float r = 0.0f, p = 0.0f, b2 = 0.0f;
        for (int w = 0; w < wavesPerBlk; ++w) { r += sR[w]; p += sP[w]; b2 += sB[w]; }
        rankPart[blockIdx.x] = r;
        npPart[blockIdx.x]   = p;
        bsPart[blockIdx.x]   = b2;
    }
}

// ---------------------------------------------------------------------------
// Kernel 3: fixed-order final combine (f64) -> scalar. Deterministic.
// ---------------------------------------------------------------------------
__global__ void k3_final(const float* __restrict__ nllPart,
                         const float* __restrict__ mPart, int nK1,
                         const float* __restrict__ rankPart,
                         const float* __restrict__ npPart,
                         const float* __restrict__ bsPart, int nK2,
                         float* __restrict__ out, int B)
{
    if (threadIdx.x == 0 && blockIdx.x == 0) {
        double nll = 0.0, m = 0.0;
        for (int k = 0; k < nK1; ++k) { nll += nllPart[k]; m += mPart[k]; }
        double rs = 0.0, np = 0.0, bs = 0.0;
        for (int k = 0; k < nK2; ++k) {
            rs += rankPart[k]; np += npPart[k]; bs += bsPart[k];
        }
        double nll_loss   = nll / (double)B;
        double rank_loss  = (np > 0.0) ? (rs / fmax(np, 1.0)) : 0.0;
        double denom      = fmax(m * (m - 1.0), 1.0);
        double bound_loss = (m > 1.0) ? (bs / denom) : 0.0;
        out[0] = (float)(nll_loss + 0.5 * rank_loss + 0.1 * bound_loss);
    }
}

// ---------------------------------------------------------------------------
extern "C" void kernel_launch(void* const* d_in, const int* in_sizes, int n_in,
                              void* d_out, int out_size, void* d_ws, size_t ws_size,
                              hipStream_t stream)
{
    const float* rp     = (const float*)d_in[0];   // [B, 64]
    const int*   times  = (const int*)  d_in[1];   // [B]
    const int*   events = (const int*)  d_in[2];   // [B]
    const float* bnd    = (const float*)d_in[3];   // [B]
    float*       out    = (float*)d_out;

    const int B   = in_sizes[1];                   // 8192
    const int nK1 = (B + 255) / 256;

    // Workspace layout
    float*  ws       = (float*)d_ws;
    float*  C        = ws;                          // B * 64 floats
    float4* P        = (float4*)(C + (size_t)B * T_DIM);   // B float4
    float4* Q        = P + B;                       // B float4
    float*  nllPart  = (float*)(Q + B);             // nK1
    float*  mPart    = nllPart + nK1;               // nK1
    float*  rankPart = mPart + nK1;                 // K2_BLOCKS
    float*  npPart   = rankPart + K2_BLOCKS;        // K2_BLOCKS
    float*  bsPart   = npPart + K2_BLOCKS;          // K2_BLOCKS

    k1_rowstats<<<nK1, 256, 0, stream>>>(rp, times, events, bnd,
                                         C, P, Q, nllPart, mPart, B);

    k2_pairwise<<<K2_BLOCKS, 256, 0, stream>>>(C, times, P, Q,
                                               rankPart, npPart, bsPart, B);

    k3_final<<<1, 32, 0, stream>>>(nllPart, mPart, nK1,
                                   rankPart, npPart, bsPart, K2_BLOCKS,
                                   out, B);
}